// GraphConv_14035953123516
// MI455X (gfx1250) — compile-verified
//
#include <hip/hip_runtime.h>

#define D 64  // D_IN == D_OUT == 64

typedef float v2f __attribute__((ext_vector_type(2)));
typedef float v8f __attribute__((ext_vector_type(8)));

// ---------------------------------------------------------------------------
// Phase 0: zero the accumulation buffer (d_out doubles as agg scratch).
// ---------------------------------------------------------------------------
__global__ void zero_f4(float4* __restrict__ p, int n4) {
  int i = blockIdx.x * blockDim.x + threadIdx.x;
  if (i < n4) p[i] = make_float4(0.f, 0.f, 0.f, 0.f);
}

// ---------------------------------------------------------------------------
// Phase 1: edge scatter-add.  16 threads per edge; each thread gathers one
// float4 (b128, 16B-aligned, coalesced) of features[src] and issues 4 native
// f32 atomics into agg[dst].  unsafeAtomicAdd forces the hardware
// global/flat_atomic_add_f32 path (no CAS loop).  features (25.6MB) and agg
// (25.6MB) are both L2-resident, so this phase is L2 atomic-throughput bound;
// 64 atomics/edge is the ISA floor (no packed-f32 atomic exists).
// ---------------------------------------------------------------------------
__global__ void scatter_add(const float4* __restrict__ feat4,
                            const int* __restrict__ src,
                            const int* __restrict__ dst,
                            float* agg, int nEdges) {
  unsigned long long tid =
      (unsigned long long)blockIdx.x * blockDim.x + threadIdx.x;
  int edge = (int)(tid >> 4);
  if (edge >= nEdges) return;
  int q = (int)(tid & 15);  // which float4 of the 64-float row
  int s = src[edge];
  int d = dst[edge];
  float4 v = feat4[(size_t)s * (D / 4) + q];
  float* ap = agg + (size_t)d * D + q * 4;
  unsafeAtomicAdd(ap + 0, v.x);
  unsafeAtomicAdd(ap + 1, v.y);
  unsafeAtomicAdd(ap + 2, v.z);
  unsafeAtomicAdd(ap + 3, v.w);
}

// ---------------------------------------------------------------------------
// Phase 2: out = agg @ W.T + b using V_WMMA_F32_16X16X4_F32 (full fp32).
// One wave32 owns 16 rows x 64 cols. In-place on d_out is safe: each wave
// preloads its entire 16x64 A tile into registers (load->wmma->store
// dependence chain) before storing to the same rows; waves own disjoint rows.
//
// f32 WMMA layouts (ISA 7.12.2):
//   A 16x4 :  lane L -> m = L%16 ;  vgpr pair = A[m][(L/16)*2 + {0,1}]
//   B 4x16 :  lane L -> n = L%16 ;  vgpr pair = B[(L/16)*2 + {0,1}][n]
//             and B[k][n] = W[n][k]  (out = agg @ W.T), so each lane reads
//             two consecutive floats from row n of W.
//   C/D    :  lane L, vgpr r -> D[(L/16)*8 + r][L%16]
// ---------------------------------------------------------------------------
__global__ void gemm_bias_wmma(const float* agg, const float* __restrict__ Wt,
                               const float* __restrict__ bias, float* out,
                               int mtiles, int nNodes) {
  int wv = (int)(((unsigned)blockIdx.x * blockDim.x + threadIdx.x) >> 5);
  if (wv >= mtiles) return;  // wave-uniform: EXEC stays all-ones for WMMA

  int lane = threadIdx.x & 31;
  int half = lane >> 4;  // 0: K={0,1} / C rows 0..7 ; 1: K={2,3} / rows 8..15
  int lm = lane & 15;
  int m0 = wv * 16;

  // Preload the full 16x64 A tile (32 VGPRs). Clamp row for ragged tails.
  int arowi = m0 + lm;
  if (arowi > nNodes - 1) arowi = nNodes - 1;
  const float* arow = agg + (size_t)arowi * D + half * 2;
  v2f a[16];
#pragma unroll
  for (int kk = 0; kk < 16; ++kk) a[kk] = *(const v2f*)(arow + kk * 4);

#pragma unroll
  for (int nt = 0; nt < 4; ++nt) {
    int col = nt * 16 + lm;
    const float* brow = Wt + (size_t)col * D + half * 2;  // W row n, k-pairs
    v8f c = {};
#pragma unroll
    for (int kk = 0; kk < 16; ++kk) {
      v2f bv = *(const v2f*)(brow + kk * 4);
      // (neg_a, A, neg_b, B, c_mod, C, reuse_a, reuse_b)
      c = __builtin_amdgcn_wmma_f32_16x16x4_f32(false, a[kk], false, bv,
                                                (short)0, c, false, false);
    }
    float bb = bias[col];
    float* op = out + (size_t)(m0 + half * 8) * D + col;
#pragma unroll
    for (int r = 0; r < 8; ++r) {
      int row = m0 + half * 8 + r;
      if (row < nNodes) op[(size_t)r * D] = c[r] + bb;
    }
  }
}

// ---------------------------------------------------------------------------
// inputs: 0=features [N,64] f32, 1=src [E] i32, 2=dst [E] i32,
//         3=W [64,64] f32, 4=b [64] f32 ; output: [N,64] f32
// ---------------------------------------------------------------------------
extern "C" void kernel_launch(void* const* d_in, const int* in_sizes, int n_in,
                              void* d_out, int out_size, void* d_ws,
                              size_t ws_size, hipStream_t stream) {
  const float* feat = (const float*)d_in[0];
  const int* src = (const int*)d_in[1];
  const int* dst = (const int*)d_in[2];
  const float* W = (const float*)d_in[3];
  const float* b = (const float*)d_in[4];
  float* out = (float*)d_out;

  int nNodes = in_sizes[0] / D;
  int nEdges = in_sizes[1];

  // Phase 0: zero accumulator (d_out) -- 1.6M float4 stores.
  int n4 = out_size / 4;
  zero_f4<<<(n4 + 255) / 256, 256, 0, stream>>>((float4*)out, n4);

  // Phase 1: scatter-add, 16 threads per edge (float4 gather + 4 atomics).
  long long work = (long long)nEdges * 16;
  int blocks = (int)((work + 255) / 256);
  scatter_add<<<blocks, 256, 0, stream>>>((const float4*)feat, src, dst, out,
                                          nEdges);

  // Phase 2: fp32 WMMA GEMM + bias, in place on d_out.
  int mtiles = (nNodes + 15) / 16;  // 6250 for N=100000
  gemm_bias_wmma<<<(mtiles + 7) / 8, 256, 0, stream>>>(out, W, b, out, mtiles,
                                                       nNodes);
}